// SynGNNLayer_27152783245552
// MI455X (gfx1250) — compile-verified
//
#include <hip/hip_runtime.h>
#include <float.h>
#include <math.h>

// ---------------- problem constants (match reference) ----------------
#define NN   50000
#define EE   400000
#define DD   64
#define HH   4
#define CC   64
#define EDD  32
#define FFF  2048
#define TOT  (EE + NN)
#define EPS  1e-5f

typedef __attribute__((ext_vector_type(2))) float v2f;
typedef __attribute__((ext_vector_type(8))) float v8f;

// f32 WMMA 16x16x4 (exact fp32 path on CDNA5), 8-arg form per probe:
// (neg_a, A, neg_b, B, c_mod, C, reuse_a, reuse_b)
__device__ __forceinline__ v8f wmma4(v2f a, v2f b, v8f c) {
  return __builtin_amdgcn_wmma_f32_16x16x4_f32(false, a, false, b, (short)0, c,
                                               false, false);
}

// order-preserving float<->uint for atomicMax on floats
__device__ __forceinline__ unsigned f2o(float f) {
  unsigned u = __float_as_uint(f);
  return (u & 0x80000000u) ? ~u : (u | 0x80000000u);
}
__device__ __forceinline__ float o2f(unsigned u) {
  return (u & 0x80000000u) ? __uint_as_float(u & 0x7fffffffu)
                           : __uint_as_float(~u);
}

// ---------------- graph-norm: global mean/var reduction ----------------
__global__ void k_reduce_sumsq(const float* __restrict__ x, int n,
                               float* __restrict__ accum) {
  __shared__ float s1[256], s2[256];
  float a = 0.f, b = 0.f;
  for (int i = blockIdx.x * blockDim.x + threadIdx.x; i < n;
       i += gridDim.x * blockDim.x) {
    float v = x[i];
    a += v;
    b += v * v;
  }
  s1[threadIdx.x] = a;
  s2[threadIdx.x] = b;
  __syncthreads();
  for (int st = 128; st > 0; st >>= 1) {
    if ((int)threadIdx.x < st) {
      s1[threadIdx.x] += s1[threadIdx.x + st];
      s2[threadIdx.x] += s2[threadIdx.x + st];
    }
    __syncthreads();
  }
  if (threadIdx.x == 0) {
    atomicAdd(&accum[0], s1[0]);
    atomicAdd(&accum[1], s2[0]);
  }
}

__global__ void k_finalize_stats(const float* __restrict__ accum, float invn,
                                 float* __restrict__ stats) {
  float mean = accum[0] * invn;
  float var  = fmaxf(accum[1] * invn - mean * mean, 0.f);
  stats[0] = mean;
  stats[1] = 1.f / (sqrtf(var) + EPS);
}

__global__ void k_apply_norm(const float* __restrict__ x,
                             const float* __restrict__ stats,
                             const float* __restrict__ w,
                             const float* __restrict__ b,
                             float* __restrict__ out, int n) {
  int i = blockIdx.x * blockDim.x + threadIdx.x;
  if (i >= n) return;
  int c = i & (DD - 1);
  out[i] = w[c] * ((x[i] - stats[0]) * stats[1]) + b[c];
}

// ---------------- edge_attr column mean -> self-loop embedding ----------------
__global__ void k_colsum(const float* __restrict__ ea, float* __restrict__ cs) {
  __shared__ float s[EDD];
  if (threadIdx.x < EDD) s[threadIdx.x] = 0.f;
  __syncthreads();
  for (int i = blockIdx.x * blockDim.x + threadIdx.x; i < EE * EDD;
       i += gridDim.x * blockDim.x)
    atomicAdd(&s[i & (EDD - 1)], ea[i]);
  __syncthreads();
  if (threadIdx.x < EDD) atomicAdd(&cs[threadIdx.x], s[threadIdx.x]);
}

__global__ void k_loop_emb(const float* __restrict__ cs,
                           const float* __restrict__ W,
                           float* __restrict__ le, float invE) {
  int c = threadIdx.x;  // 256 threads
  float acc = 0.f;
#pragma unroll
  for (int k = 0; k < EDD; ++k) acc = fmaf(cs[k] * invE, W[k * 256 + c], acc);
  le[c] = acc;
}

// ---------------- WMMA GEMM: C[M x 256] = A[M x 64] @ B[64 x 256] + bias -----
// block = 256 threads (8 wave32), M-tile 64, N-tile 128 (blockIdx.y in {0,1})
__global__ void k_gemm64_wmma(const float* __restrict__ A,
                              const float* __restrict__ B,
                              const float* __restrict__ bias,
                              float* __restrict__ C, int M) {
  __shared__ float As[64][68];
  const int wid = threadIdx.x >> 5, lane = threadIdx.x & 31;
  const int row0 = blockIdx.x * 64, col0 = blockIdx.y * 128;
  for (int i = threadIdx.x; i < 64 * 64; i += 256) {
    int r = i >> 6, c = i & 63, gr = row0 + r;
    As[r][c] = (gr < M) ? A[(size_t)gr * 64 + c] : 0.f;
  }
  __syncthreads();
  const int l16 = lane & 15, kh = (lane >> 4) * 2, vh = (lane >> 4) * 8;
  for (int t = 0; t < 4; ++t) {
    int tile = wid * 4 + t;
    int mt = tile >> 3, nt = tile & 7;
    int col = col0 + nt * 16 + l16;
    v8f acc;
#pragma unroll
    for (int v = 0; v < 8; ++v) acc[v] = 0.f;
#pragma unroll
    for (int k = 0; k < 64; k += 4) {
      v2f a, b;
      a.x = As[mt * 16 + l16][k + kh];
      a.y = As[mt * 16 + l16][k + kh + 1];
      b.x = B[(size_t)(k + kh) * 256 + col];
      b.y = B[(size_t)(k + kh + 1) * 256 + col];
      acc = wmma4(a, b, acc);
    }
    float bv = bias[col];
#pragma unroll
    for (int v = 0; v < 8; ++v) {
      int gr = row0 + mt * 16 + vh + v;
      if (gr < M) C[(size_t)gr * 256 + col] = acc[v] + bv;
    }
  }
}

// ---------------- per-edge GATv2 logits (wave per edge) ----------------
// e_emb recomputed on the fly from 32KB L2-resident lin_edge_w (saves 400MB HBM)
__global__ void k_edge_logits(const float* __restrict__ xl,
                              const float* __restrict__ xr,
                              const float* __restrict__ ea,
                              const float* __restrict__ lew,
                              const float* __restrict__ lemb,
                              const float* __restrict__ att,
                              const int* __restrict__ ei,
                              float* __restrict__ lg) {
  int gw = (blockIdx.x * blockDim.x + threadIdx.x) >> 5;  // edge id
  int lane = threadIdx.x & 31;
  if (gw >= TOT) return;
  int s, d, ch0 = lane * 8;
  float eemb[8];
  if (gw < EE) {
    s = ei[gw];
    d = ei[EE + gw];
    float myk = ea[(size_t)gw * EDD + lane];
#pragma unroll
    for (int j = 0; j < 8; ++j) eemb[j] = 0.f;
    for (int k = 0; k < EDD; ++k) {
      float a = __shfl(myk, k, 32);
      const float* wr = lew + k * 256 + ch0;
#pragma unroll
      for (int j = 0; j < 8; ++j) eemb[j] = fmaf(a, wr[j], eemb[j]);
    }
  } else {
    s = d = gw - EE;
#pragma unroll
    for (int j = 0; j < 8; ++j) eemb[j] = lemb[ch0 + j];
  }
  const float* xs = xl + (size_t)s * 256 + ch0;
  const float* xd = xr + (size_t)d * 256 + ch0;
  const float* aw = att + ch0;  // att flat [H*C] == channel index
  float dot = 0.f;
#pragma unroll
  for (int j = 0; j < 8; ++j) {
    float g = xs[j] + xd[j] + eemb[j];
    g = (g >= 0.f) ? g : 0.2f * g;  // LeakyReLU(0.2)
    dot = fmaf(g, aw[j], dot);
  }
  dot += __shfl_xor(dot, 1, 32);
  dot += __shfl_xor(dot, 2, 32);
  dot += __shfl_xor(dot, 4, 32);
  if ((lane & 7) == 0) lg[(size_t)gw * HH + (lane >> 3)] = dot;
}

// ---------------- segment softmax ----------------
__global__ void k_init_m(unsigned* __restrict__ m, int n) {
  int i = blockIdx.x * blockDim.x + threadIdx.x;
  if (i < n) m[i] = f2o(-FLT_MAX);
}

__global__ void k_segmax(const float* __restrict__ lg, const int* __restrict__ ei,
                         unsigned* __restrict__ m, int cnt) {
  int i = blockIdx.x * blockDim.x + threadIdx.x;
  if (i >= cnt) return;
  int e = i >> 2, h = i & 3;
  int d = (e < EE) ? ei[EE + e] : (e - EE);
  atomicMax(&m[d * HH + h], f2o(lg[i]));
}

__global__ void k_exp_sum(float* __restrict__ lg, const int* __restrict__ ei,
                          const unsigned* __restrict__ m,
                          float* __restrict__ den, int cnt) {
  int i = blockIdx.x * blockDim.x + threadIdx.x;
  if (i >= cnt) return;
  int e = i >> 2, h = i & 3;
  int d = (e < EE) ? ei[EE + e] : (e - EE);
  float ex = expf(lg[i] - o2f(m[d * HH + h]));
  lg[i] = ex;  // in-place: logits -> exp
  atomicAdd(&den[d * HH + h], ex);
}

// ---------------- alpha + scatter aggregation (wave per edge) ----------------
__global__ void k_alpha_agg(const float* __restrict__ ex,
                            const float* __restrict__ den,
                            const float* __restrict__ xl,
                            const int* __restrict__ ei,
                            float* __restrict__ alpha_out,
                            float* __restrict__ agg) {
  int gw = (blockIdx.x * blockDim.x + threadIdx.x) >> 5;
  int lane = threadIdx.x & 31;
  if (gw >= TOT) return;
  int s, d;
  if (gw < EE) { s = ei[gw]; d = ei[EE + gw]; } else { s = d = gw - EE; }
  int h = lane >> 3;
  float al = ex[(size_t)gw * HH + h] / den[d * HH + h];
  if (lane < HH)
    alpha_out[(size_t)gw * HH + lane] =
        ex[(size_t)gw * HH + lane] / den[d * HH + lane];
  int ch0 = lane * 8;
  const float* xs = xl + (size_t)s * 256 + ch0;
  float* ag = agg + (size_t)d * 256 + ch0;
#pragma unroll
  for (int j = 0; j < 8; ++j) atomicAdd(&ag[j], al * xs[j]);
}

__global__ void k_head_mean_residual(const float* __restrict__ src,
                                     const float* __restrict__ agg,
                                     const float* __restrict__ gbias,
                                     float* __restrict__ h1, int n) {
  int i = blockIdx.x * blockDim.x + threadIdx.x;
  if (i >= n) return;
  int node = i >> 6, c = i & 63;
  const float* a = agg + (size_t)node * 256 + c;
  float mh = 0.25f * (a[0] + a[64] + a[128] + a[192]);
  h1[i] = src[i] + mh + gbias[c];
}

// ---------------- fused FFN: out = h1 + gelu(hn@w1+b1)@w2 + b2 --------------
// 64-row tile; 2048-wide activation lives only in LDS (128-wide chunks).
__global__ void k_ffn_fused(const float* __restrict__ hn,
                            const float* __restrict__ h1,
                            const float* __restrict__ w1,
                            const float* __restrict__ b1,
                            const float* __restrict__ w2,
                            const float* __restrict__ b2,
                            float* __restrict__ out, int M) {
  __shared__ float Hs[64][68];
  __shared__ float Act[64][132];
  const int wid = threadIdx.x >> 5, lane = threadIdx.x & 31;
  const int row0 = blockIdx.x * 64;
  for (int i = threadIdx.x; i < 64 * 64; i += 256) {
    int r = i >> 6, c = i & 63, gr = row0 + r;
    Hs[r][c] = (gr < M) ? hn[(size_t)gr * 64 + c] : 0.f;
  }
  __syncthreads();
  const int l16 = lane & 15, kh = (lane >> 4) * 2, vh = (lane >> 4) * 8;
  v8f acc2[2];
#pragma unroll
  for (int t = 0; t < 2; ++t)
#pragma unroll
    for (int v = 0; v < 8; ++v) acc2[t][v] = 0.f;

  for (int f = 0; f < FFF; f += 128) {
    // GEMM1: 64x128 chunk = gelu(Hs @ w1[:, f:f+128] + b1)
    for (int t = 0; t < 4; ++t) {
      int tile = wid * 4 + t;
      int mt = tile >> 3, nt = tile & 7;
      int col = f + nt * 16 + l16;
      v8f a1;
#pragma unroll
      for (int v = 0; v < 8; ++v) a1[v] = 0.f;
#pragma unroll
      for (int k = 0; k < 64; k += 4) {
        v2f a, b;
        a.x = Hs[mt * 16 + l16][k + kh];
        a.y = Hs[mt * 16 + l16][k + kh + 1];
        b.x = w1[(size_t)(k + kh) * FFF + col];
        b.y = w1[(size_t)(k + kh + 1) * FFF + col];
        a1 = wmma4(a, b, a1);
      }
      float bv = b1[col];
#pragma unroll
      for (int v = 0; v < 8; ++v) {
        float xv = a1[v] + bv;
        float g = 0.5f * xv * (1.f + erff(xv * 0.7071067811865475f));  // exact gelu
        Act[mt * 16 + vh + v][nt * 16 + l16] = g;
      }
    }
    __syncthreads();
    // GEMM2: out_acc(64x64) += Act(64x128) @ w2[f:f+128, :]
    for (int t = 0; t < 2; ++t) {
      int tile = wid * 2 + t;
      int mt = tile >> 2, nt = tile & 3;
      int col = nt * 16 + l16;
#pragma unroll
      for (int k = 0; k < 128; k += 4) {
        v2f a, b;
        a.x = Act[mt * 16 + l16][k + kh];
        a.y = Act[mt * 16 + l16][k + kh + 1];
        b.x = w2[(size_t)(f + k + kh) * 64 + col];
        b.y = w2[(size_t)(f + k + kh + 1) * 64 + col];
        acc2[t] = wmma4(a, b, acc2[t]);
      }
    }
    __syncthreads();
  }
  for (int t = 0; t < 2; ++t) {
    int tile = wid * 2 + t;
    int mt = tile >> 2, nt = tile & 3;
    int col = nt * 16 + l16;
#pragma unroll
    for (int v = 0; v < 8; ++v) {
      int gr = row0 + mt * 16 + vh + v;
      if (gr < M)
        out[(size_t)gr * 64 + col] =
            acc2[t][v] + b2[col] + h1[(size_t)gr * 64 + col];
    }
  }
}

// ---------------- host orchestration ----------------
extern "C" void kernel_launch(void* const* d_in, const int* in_sizes, int n_in,
                              void* d_out, int out_size, void* d_ws,
                              size_t ws_size, hipStream_t stream) {
  const float* x    = (const float*)d_in[0];
  const int*   ei   = (const int*)d_in[1];
  const float* ea   = (const float*)d_in[2];
  // d_in[3] = batch (unused; all zeros)
  const float* llw  = (const float*)d_in[4];
  const float* llb  = (const float*)d_in[5];
  const float* lrw  = (const float*)d_in[6];
  const float* lrb  = (const float*)d_in[7];
  const float* lew  = (const float*)d_in[8];
  const float* att  = (const float*)d_in[9];
  const float* gb   = (const float*)d_in[10];
  const float* w1   = (const float*)d_in[11];
  const float* b1   = (const float*)d_in[12];
  const float* w2   = (const float*)d_in[13];
  const float* b2   = (const float*)d_in[14];
  const float* n0w  = (const float*)d_in[15];
  const float* n0b  = (const float*)d_in[16];
  const float* n1w  = (const float*)d_in[17];
  const float* n1b  = (const float*)d_in[18];
  const float* n2w  = (const float*)d_in[19];
  const float* n2b  = (const float*)d_in[20];

  float* out       = (float*)d_out;             // [N*64] h, then [TOT*4] alpha
  float* alpha_out = out + (size_t)NN * 64;

  // workspace layout (~137 MB with reuse)
  float* ws   = (float*)d_ws;
  float* src  = ws;                                // N*64   (reused as hn later)
  float* xl   = src + (size_t)NN * 64;             // N*256  (reused as h2 later)
  float* xr   = xl + (size_t)NN * 256;             // N*256  (reused as agg later)
  float* lg   = xr + (size_t)NN * 256;             // TOT*4  logits -> exp
  float* mf   = lg + (size_t)TOT * 4;              // N*4    (as ordered-uint)
  float* den  = mf + (size_t)NN * 4;               // N*4
  float* h1   = den + (size_t)NN * 4;              // N*64
  float* cs   = h1 + (size_t)NN * 64;              // 32
  float* lemb = cs + 32;                           // 256
  float* acm  = lemb + 256;                        // 2
  float* sts  = acm + 2;                           // 2
  unsigned* mbuf = (unsigned*)mf;
  float* agg = xr;   // xr dead after logits
  float* hn  = src;  // src dead after residual
  float* h2  = xl;   // xl dead after aggregation

  const int ND = NN * DD;

  // ---- graph_norm 0 : x -> src ----
  hipMemsetAsync(acm, 0, 2 * sizeof(float), stream);
  k_reduce_sumsq<<<512, 256, 0, stream>>>(x, ND, acm);
  k_finalize_stats<<<1, 1, 0, stream>>>(acm, 1.f / ND, sts);
  k_apply_norm<<<(ND + 255) / 256, 256, 0, stream>>>(x, sts, n0w, n0b, src, ND);

  // ---- self-loop edge embedding ----
  hipMemsetAsync(cs, 0, EDD * sizeof(float), stream);
  k_colsum<<<1024, 256, 0, stream>>>(ea, cs);
  k_loop_emb<<<1, 256, 0, stream>>>(cs, lew, lemb, 1.f / EE);

  // ---- x_l / x_r projections (WMMA) ----
  dim3 ggrid((NN + 63) / 64, 2);
  k_gemm64_wmma<<<ggrid, 256, 0, stream>>>(src, llw, llb, xl, NN);
  k_gemm64_wmma<<<ggrid, 256, 0, stream>>>(src, lrw, lrb, xr, NN);

  // ---- per-edge logits ----
  k_edge_logits<<<(TOT + 7) / 8, 256, 0, stream>>>(xl, xr, ea, lew, lemb, att,
                                                   ei, lg);

  // ---- segment softmax ----
  k_init_m<<<(NN * HH + 255) / 256, 256, 0, stream>>>(mbuf, NN * HH);
  hipMemsetAsync(den, 0, (size_t)NN * HH * sizeof(float), stream);
  int cnt = TOT * HH;
  k_segmax<<<(cnt + 255) / 256, 256, 0, stream>>>(lg, ei, mbuf, cnt);
  k_exp_sum<<<(cnt + 255) / 256, 256, 0, stream>>>(lg, ei, mbuf, den, cnt);

  // ---- alpha + aggregation ----
  hipMemsetAsync(agg, 0, (size_t)NN * 256 * sizeof(float), stream);
  k_alpha_agg<<<(TOT + 7) / 8, 256, 0, stream>>>(lg, den, xl, ei, alpha_out,
                                                 agg);
  k_head_mean_residual<<<(ND + 255) / 256, 256, 0, stream>>>(src, agg, gb, h1,
                                                             ND);

  // ---- graph_norm 1 : h1 -> hn ----
  hipMemsetAsync(acm, 0, 2 * sizeof(float), stream);
  k_reduce_sumsq<<<512, 256, 0, stream>>>(h1, ND, acm);
  k_finalize_stats<<<1, 1, 0, stream>>>(acm, 1.f / ND, sts);
  k_apply_norm<<<(ND + 255) / 256, 256, 0, stream>>>(h1, sts, n1w, n1b, hn, ND);

  // ---- fused FFN (WMMA, LDS-resident activations) ----
  k_ffn_fused<<<(NN + 63) / 64, 256, 0, stream>>>(hn, h1, w1, b1, w2, b2, h2,
                                                  NN);

  // ---- graph_norm 2 : h2 -> d_out ----
  hipMemsetAsync(acm, 0, 2 * sizeof(float), stream);
  k_reduce_sumsq<<<512, 256, 0, stream>>>(h2, ND, acm);
  k_finalize_stats<<<1, 1, 0, stream>>>(acm, 1.f / ND, sts);
  k_apply_norm<<<(ND + 255) / 256, 256, 0, stream>>>(h2, sts, n2w, n2b, out,
                                                     ND);
}